// HybridNATModel_65481071409227
// MI455X (gfx1250) — compile-verified
//
#include <hip/hip_runtime.h>

typedef float v2f __attribute__((ext_vector_type(2)));
typedef float v8f __attribute__((ext_vector_type(8)));

// ---- workspace layout (floats) ----
#define WS_UR    0     // Ur[16][16] row-major (row = state index j, col = k)
#define WS_UI    256   // Ui[16][16]
#define WS_P     512   // P[16][4] = ZMAT @ W^T
#define WS_SCALE 584   // scale[4] = gamma * invstd
#define WS_SHIFT 588   // shift[4] = beta - mean*scale
#define WS_PART  640   // per-block partials: nBlocks * 8 (sum[4], sumsq[4])

#define NQ 4
#define BN_EPS 1e-5f

// ---- raw cross-lane helpers (wave32, EXEC all-ones at use sites) ----
__device__ __forceinline__ float bperm_f(int srcLane, float v) {
  return __builtin_bit_cast(
      float, __builtin_amdgcn_ds_bpermute(srcLane << 2, __builtin_bit_cast(int, v)));
}
__device__ __forceinline__ float swz_xor1_f(float v) {
  // ds_swizzle group-of-32: and=0x1f, or=0, xor=1 -> offset 0x041f
  return __builtin_bit_cast(
      float, __builtin_amdgcn_ds_swizzle(__builtin_bit_cast(int, v), 0x041f));
}

// -------------------------------------------------------------------------
// Setup: build Ur/Ui (shared RY+CZ block folded with (-i)^popcount masks)
// and P = ZMAT @ W^T. One wave; thread t owns column k=t of U.
// -------------------------------------------------------------------------
__global__ void qnat_setup(const float* __restrict__ weights,
                           const float* __restrict__ W,
                           float* __restrict__ ws) {
  const int t = threadIdx.x;
  if (t >= 16) return;

  float u[16];
#pragma unroll
  for (int j = 0; j < 16; ++j) u[j] = (j == t) ? 1.0f : 0.0f;

  int idx = 0;
#pragma unroll
  for (int layer = 0; layer < 2; ++layer) {
    // RY(weights[idx]) on wires 0..3 ; wire w acts on bit (NQ-1-w)
#pragma unroll
    for (int w = 0; w < 4; ++w) {
      const float th = weights[idx++] * 0.5f;
      const float c = __cosf(th), s = __sinf(th);
      const int bit = 1 << (NQ - 1 - w);
#pragma unroll
      for (int j = 0; j < 16; ++j) {
        if (!(j & bit)) {
          const int j1 = j | bit;
          const float a0 = u[j], a1 = u[j1];
          u[j]  = c * a0 - s * a1;   // g = [[c,-s],[s,c]]
          u[j1] = s * a0 + c * a1;
        }
      }
    }
    // CZ chain on (0,1),(1,2),(2,3) -> diagonal sign on adjacent set bits
#pragma unroll
    for (int j = 0; j < 16; ++j) {
      float sgn = 1.0f;
      if ((j & 8) && (j & 4)) sgn = -sgn;
      if ((j & 4) && (j & 2)) sgn = -sgn;
      if ((j & 2) && (j & 1)) sgn = -sgn;
      u[j] *= sgn;
    }
  }

  // psi[k] = m[k] * (-i)^popcount(k): real mask rho, imag mask sigma
  const int pc = __popc(t) & 3;
  const float rho = (pc == 0) ? 1.0f : ((pc == 2) ? -1.0f : 0.0f);
  const float sig = (pc == 1) ? -1.0f : ((pc == 3) ? 1.0f : 0.0f);
#pragma unroll
  for (int j = 0; j < 16; ++j) {
    ws[WS_UR + j * 16 + t] = u[j] * rho;
    ws[WS_UI + j * 16 + t] = u[j] * sig;
  }

  // P[k][c] = sum_w ZMAT[k][w] * W[c][w]
#pragma unroll
  for (int c = 0; c < 4; ++c) {
    float acc = 0.0f;
#pragma unroll
    for (int w = 0; w < 4; ++w) {
      const float z = ((t >> (NQ - 1 - w)) & 1) ? -1.0f : 1.0f;
      acc += z * W[c * 4 + w];
    }
    ws[WS_P + t * 4 + c] = acc;
  }
}

// -------------------------------------------------------------------------
// Main: pooling + RX product-state + WMMA 16x16 transform + logits.
// 256 threads = 8 waves; each wave does 8 tiles of 16 batch rows.
// Writes raw logits (no bias; it cancels under BN) to out, and
// deterministic per-block [sum[4], sumsq[4]] partials to `part`.
// -------------------------------------------------------------------------
#define TPAD 20  // row stride (floats) of the LDS transpose tile; 80B keeps 16B align

__global__ __launch_bounds__(256) void qnat_main(const float* __restrict__ x,
                                                 const float* __restrict__ ws,
                                                 float* __restrict__ out,
                                                 float* __restrict__ part,
                                                 int B) {
  __shared__ float smem[8 * 16 * TPAD];  // per-wave transpose tiles; reused for BN

  const int lane = threadIdx.x & 31;
  const int wave = threadIdx.x >> 5;
  const int half = lane >> 4;     // 0: lanes 0-15, 1: lanes 16-31
  const int nIdx = lane & 15;     // N index (state j) for B/C/D operands
  const int koff = half ? 2 : 0;  // K offset within a K=4 slice (A & B layout)
  float* const sT = smem + wave * 16 * TPAD;

  // B operands: slice t holds K = 4t+koff, 4t+koff+1 for column n = nIdx
  v2f br[4], bi[4];
#pragma unroll
  for (int t = 0; t < 4; ++t) {
    br[t] = *(const v2f*)(ws + WS_UR + nIdx * 16 + 4 * t + koff);
    bi[t] = *(const v2f*)(ws + WS_UI + nIdx * 16 + 4 * t + koff);
  }

  // This lane's half of the probs->logits projection: rows j0..j0+7 of P
  const int row = lane >> 1;       // batch row this lane finalizes (0..15)
  const int j0  = (lane & 1) * 8;  // which 8 states this lane dots over
  float4 Pv[8];
#pragma unroll
  for (int k = 0; k < 8; ++k) Pv[k] = *(const float4*)(ws + WS_P + (j0 + k) * 4);

  float rs0 = 0.0f, rs1 = 0.0f, rq0 = 0.0f, rq1 = 0.0f;  // BN running sums

  for (int tt = 0; tt < 8; ++tt) {
    const int tile = blockIdx.x * 64 + wave + 8 * tt;
    const int rowBase = tile * 16;
    if (rowBase < B) {
      // ---- 6x6 avg-pool: 2 lanes per image; each lane does 6 rows ----
      const float* px = x + (size_t)(rowBase + (lane >> 1)) * 144
                          + (size_t)(lane & 1) * 72;
      float sL = 0.0f, sR = 0.0f;
#pragma unroll
      for (int rr = 0; rr < 6; ++rr) {
        const float4 v0 = *(const float4*)(px + rr * 12 + 0);
        const float4 v1 = *(const float4*)(px + rr * 12 + 4);
        const float4 v2 = *(const float4*)(px + rr * 12 + 8);
        sL += v0.x + v0.y + v0.z + v0.w + v1.x + v1.y;
        sR += v1.z + v1.w + v2.x + v2.y + v2.z + v2.w;
      }
      sL *= (1.0f / 36.0f);
      sR *= (1.0f / 36.0f);

      // gather the 4 pooled angles of item nIdx (raw bpermute, no bounds check)
      const int src = nIdx * 2;
      const float p0 = bperm_f(src, sL);
      const float p1 = bperm_f(src, sR);
      const float p2 = bperm_f(src + 1, sL);
      const float p3 = bperm_f(src + 1, sR);

      float c0, s0, c1, s1, c2, s2, c3, s3;
      __sincosf(p0 * 0.5f, &s0, &c0);
      __sincosf(p1 * 0.5f, &s1, &c1);
      __sincosf(p2 * 0.5f, &s2, &c2);
      __sincosf(p3 * 0.5f, &s3, &c3);

      // m[k] = hi[k>>2] * lo[k&3]  (bit3->w0, bit2->w1, bit1->w2, bit0->w3)
      const float hi0 = c0 * c1, hi1 = c0 * s1, hi2 = s0 * c1, hi3 = s0 * s1;
      const float lo0 = c2 * c3, lo1 = c2 * s3, lo2 = s2 * c3, lo3 = s2 * s3;
      const float loA = half ? lo2 : lo0;  // lo[koff]
      const float loB = half ? lo3 : lo1;  // lo[koff+1]

      v2f a[4];
      a[0] = (v2f){hi0 * loA, hi0 * loB};
      a[1] = (v2f){hi1 * loA, hi1 * loB};
      a[2] = (v2f){hi2 * loA, hi2 * loB};
      a[3] = (v2f){hi3 * loA, hi3 * loB};

      // ---- WMMA: final_real/imag[16 rows x 16 states], fp32 exact ----
      v8f dr = {0.f, 0.f, 0.f, 0.f, 0.f, 0.f, 0.f, 0.f};
      v8f di = {0.f, 0.f, 0.f, 0.f, 0.f, 0.f, 0.f, 0.f};
#pragma unroll
      for (int t = 0; t < 4; ++t)
        dr = __builtin_amdgcn_wmma_f32_16x16x4_f32(false, a[t], false, br[t],
                                                   (short)0, dr, false, false);
#pragma unroll
      for (int t = 0; t < 4; ++t)
        di = __builtin_amdgcn_wmma_f32_16x16x4_f32(false, a[t], false, bi[t],
                                                   (short)0, di, false, false);

      // probs -> LDS transpose tile (column j=nIdx, rows r+8*half).
      // Same-wave DS ops are processed in-order: stores complete before loads.
#pragma unroll
      for (int r = 0; r < 8; ++r) {
        const float p = dr[r] * dr[r] + di[r] * di[r];
        sT[(r + 8 * half) * TPAD + nIdx] = p;
      }

      // read back row-major: this lane dots 8 states of its batch row
      const float4 qa = *(const float4*)(sT + row * TPAD + j0);
      const float4 qb = *(const float4*)(sT + row * TPAD + j0 + 4);
      const float qv[8] = {qa.x, qa.y, qa.z, qa.w, qb.x, qb.y, qb.z, qb.w};
      float l0 = 0.f, l1 = 0.f, l2 = 0.f, l3 = 0.f;
#pragma unroll
      for (int k = 0; k < 8; ++k) {
        l0 = fmaf(qv[k], Pv[k].x, l0);
        l1 = fmaf(qv[k], Pv[k].y, l1);
        l2 = fmaf(qv[k], Pv[k].z, l2);
        l3 = fmaf(qv[k], Pv[k].w, l3);
      }
      // combine the two j-halves (lane pair differs only in bit 0)
      l0 += swz_xor1_f(l0);
      l1 += swz_xor1_f(l1);
      l2 += swz_xor1_f(l2);
      l3 += swz_xor1_f(l3);

      // write this lane's (row, class-pair); bias omitted (cancels in BN)
      const bool b0 = (lane & 1) != 0;
      const float wa = b0 ? l2 : l0;
      const float wb = b0 ? l3 : l1;
      float2 wv;
      wv.x = wa;
      wv.y = wb;
      *(float2*)(out + (size_t)(rowBase + row) * 4 + (lane & 1) * 2) = wv;

      // BN running sums: each (row, class) counted exactly once across lanes
      rs0 += wa;
      rs1 += wb;
      rq0 += wa * wa;
      rq1 += wb * wb;
    }
  }

  // ---- deterministic per-block BN partials (fixed-order LDS reduction) ----
  __syncthreads();  // all waves done with their transpose tiles
  smem[threadIdx.x * 4 + 0] = rs0;  // class (tid&1)*2
  smem[threadIdx.x * 4 + 1] = rs1;  // class (tid&1)*2 + 1
  smem[threadIdx.x * 4 + 2] = rq0;
  smem[threadIdx.x * 4 + 3] = rq1;
  __syncthreads();
  if (threadIdx.x < 8) {
    const int c = threadIdx.x & 3;
    const int bit = c >> 1;                               // lane parity owning class c
    const int slot = (c & 1) + (threadIdx.x >> 2) * 2;    // rs vs rq
    float s = 0.0f;
    for (int i = 0; i < 256; ++i)
      if ((i & 1) == bit) s += smem[i * 4 + slot];
    part[blockIdx.x * 8 + threadIdx.x] = s;
  }
}

// -------------------------------------------------------------------------
// BN stats: deterministic fixed-order reduction over block partials.
// -------------------------------------------------------------------------
__global__ void qnat_stats(float* __restrict__ ws,
                           const float* __restrict__ gamma,
                           const float* __restrict__ beta,
                           int nBlocks, float invB) {
  __shared__ float tot[8];
  const int t = threadIdx.x;
  if (t < 8) {
    float s = 0.0f;
    for (int i = 0; i < nBlocks; ++i) s += ws[WS_PART + i * 8 + t];
    tot[t] = s;
  }
  __syncthreads();
  if (t < 4) {
    const float mean = tot[t] * invB;
    const float var = tot[4 + t] * invB - mean * mean;
    const float sc = gamma[t] * rsqrtf(var + BN_EPS);
    ws[WS_SCALE + t] = sc;
    ws[WS_SHIFT + t] = beta[t] - mean * sc;
  }
}

// -------------------------------------------------------------------------
// In-place BN apply: one float4 row per thread.
// -------------------------------------------------------------------------
__global__ __launch_bounds__(256) void qnat_bn(float* __restrict__ out,
                                               const float* __restrict__ ws,
                                               int B) {
  const int i = blockIdx.x * blockDim.x + threadIdx.x;
  if (i < B) {
    float4 v = *(float4*)(out + (size_t)i * 4);
    const float4 sc = *(const float4*)(ws + WS_SCALE);
    const float4 sh = *(const float4*)(ws + WS_SHIFT);
    v.x = v.x * sc.x + sh.x;
    v.y = v.y * sc.y + sh.y;
    v.z = v.z * sc.z + sh.z;
    v.w = v.w * sc.w + sh.w;
    *(float4*)(out + (size_t)i * 4) = v;
  }
}

extern "C" void kernel_launch(void* const* d_in, const int* in_sizes, int n_in,
                              void* d_out, int out_size, void* d_ws, size_t ws_size,
                              hipStream_t stream) {
  const float* x       = (const float*)d_in[0];
  const float* weights = (const float*)d_in[1];
  const float* W       = (const float*)d_in[2];
  // d_in[3] = b : provably cancels under BatchNorm (per-class constant shift)
  const float* gamma   = (const float*)d_in[4];
  const float* beta    = (const float*)d_in[5];
  float* out = (float*)d_out;
  float* ws  = (float*)d_ws;

  const int B = in_sizes[0] / 144;          // [B,1,12,12]
  const int nBlocks = (B + 1023) / 1024;    // 1024 rows (64 tiles) per block

  qnat_setup<<<1, 32, 0, stream>>>(weights, W, ws);
  qnat_main<<<nBlocks, 256, 0, stream>>>(x, ws, out, ws + WS_PART, B);
  qnat_stats<<<1, 32, 0, stream>>>(ws, gamma, beta, nBlocks, 1.0f / (float)B);
  qnat_bn<<<(B + 255) / 256, 256, 0, stream>>>(out, ws, B);
}